// GAT_40261023432783
// MI455X (gfx1250) — compile-verified
//
#include <hip/hip_runtime.h>
#include <math.h>

// Problem constants (from reference)
#define NN 100000
#define EE 1600000
#define ET (EE + NN)   // edges + self loops
#define C  32          // HID (== warpSize on gfx1250)

typedef __attribute__((ext_vector_type(2))) float v2f;
typedef __attribute__((ext_vector_type(8))) float v8f;

// ---------------------------------------------------------------------------
// Dense GEMM  H[N,32] = X[N,K] @ W[K,32]   via V_WMMA_F32_16X16X4_F32.
// One wave computes a 16(M) x 32(N) tile with two 16x16 accumulators.
// A frag (16x4 f32): lanes 0-15 -> M=0..15 / K={0,1}; lanes 16-31 -> K={2,3}.
// B frag (4x16 f32): VGPR0 holds K=0 (lanes0-15) / K=2 (lanes16-31); VGPR1 K=1/K=3.
// C/D (16x16 f32): VGPR r, lane l -> (M = r + 8*(l>=16), N = l%16).
// ---------------------------------------------------------------------------
template <int K>
__global__ void gemm_wmma_f32(const float* __restrict__ X,
                              const float* __restrict__ W,
                              float* __restrict__ H) {
  const int lane = threadIdx.x & 31;
  const int wid  = (blockIdx.x * blockDim.x + threadIdx.x) >> 5;
  if (wid >= NN / 16) return;
  const int m0   = wid * 16;
  const int half = lane >> 4;     // 0 or 1
  const int l16  = lane & 15;

  const float* xr = X + (size_t)(m0 + l16) * K;   // A row for this lane
  v8f acc0 = {};
  v8f acc1 = {};

#pragma unroll
  for (int k = 0; k < K; k += 4) {
    const int ka = k + 2 * half;
    v2f a;
    a.x = xr[ka];
    a.y = xr[ka + 1];
    v2f b0, b1;
    b0.x = W[(size_t)ka * C + l16];
    b0.y = W[(size_t)(ka + 1) * C + l16];
    b1.x = W[(size_t)ka * C + 16 + l16];
    b1.y = W[(size_t)(ka + 1) * C + 16 + l16];
    acc0 = __builtin_amdgcn_wmma_f32_16x16x4_f32(false, a, false, b0,
                                                 (short)0, acc0, false, false);
    acc1 = __builtin_amdgcn_wmma_f32_16x16x4_f32(false, a, false, b1,
                                                 (short)0, acc1, false, false);
  }

#pragma unroll
  for (int r = 0; r < 8; ++r) {
    const int m = m0 + r + 8 * half;
    H[(size_t)m * C + l16]      = acc0[r];
    H[(size_t)m * C + 16 + l16] = acc1[r];
  }
}

// ---------------------------------------------------------------------------
// Per-node attention coefficients: asrc[n] = h[n,:].a_src, adst[n] = h[n,:].a_dst
// One wave per node; lane == channel; wave32 butterfly reduction.
// ---------------------------------------------------------------------------
__global__ void node_attn(const float* __restrict__ H,
                          const float* __restrict__ a_s,
                          const float* __restrict__ a_d,
                          float* __restrict__ asrc,
                          float* __restrict__ adst) {
  const int lane = threadIdx.x & 31;
  const int n    = (blockIdx.x * blockDim.x + threadIdx.x) >> 5;
  if (n >= NN) return;
  const float hv = H[(size_t)n * C + lane];
  float vs = hv * a_s[lane];
  float vd = hv * a_d[lane];
#pragma unroll
  for (int off = 16; off > 0; off >>= 1) {
    vs += __shfl_xor(vs, off, 32);
    vd += __shfl_xor(vd, off, 32);
  }
  if (lane == 0) {
    asrc[n] = vs;
    adst[n] = vd;
  }
}

// ---------------------------------------------------------------------------
// Per-layer init: m = -inf, ssum = 0, agg = 0
// ---------------------------------------------------------------------------
__global__ void init_layer(float* __restrict__ mmax, float* __restrict__ ssum,
                           float* __restrict__ agg, int aggcount) {
  const int i = blockIdx.x * blockDim.x + threadIdx.x;
  if (i < NN) {
    mmax[i] = -INFINITY;
    ssum[i] = 0.0f;
  }
  if (i < aggcount) agg[i] = 0.0f;
}

// Sign-aware float atomic max (IEEE total order via int/uint monotonicity).
__device__ inline void atomicMaxF32(float* addr, float v) {
  if (v >= 0.0f)
    atomicMax((int*)addr, __float_as_int(v));
  else
    atomicMin((unsigned int*)addr, (unsigned int)__float_as_int(v));
}

__device__ inline void edge_ids(int e, const int* __restrict__ src,
                                const int* __restrict__ dst, int& s, int& d) {
  if (e < EE) {
    s = src[e];
    d = dst[e];
  } else {                        // appended self loops
    s = d = e - EE;
  }
}

// Pass 1: e = leaky_relu(asrc[s] + adst[d]); store logit; segment max over dst.
__global__ void edge_logit_max(const int* __restrict__ src,
                               const int* __restrict__ dst,
                               const float* __restrict__ asrc,
                               const float* __restrict__ adst,
                               float* __restrict__ elog,
                               float* __restrict__ mmax) {
  const int e = blockIdx.x * blockDim.x + threadIdx.x;
  if (e >= ET) return;
  int s, d;
  edge_ids(e, src, dst, s, d);
  float v = asrc[s] + adst[d];
  v = (v > 0.0f) ? v : 0.2f * v;  // NEG_SLOPE = 0.2
  elog[e] = v;
  atomicMaxF32(&mmax[d], v);
}

// Pass 2: p = exp(e - m[d]); store p in-place; segment sum over dst.
__global__ void edge_exp_sum(const int* __restrict__ src,
                             const int* __restrict__ dst,
                             float* __restrict__ elog,
                             const float* __restrict__ mmax,
                             float* __restrict__ ssum) {
  const int e = blockIdx.x * blockDim.x + threadIdx.x;
  if (e >= ET) return;
  int s, d;
  edge_ids(e, src, dst, s, d);
  const float p = expf(elog[e] - mmax[d]);
  elog[e] = p;
  atomicAdd(&ssum[d], p);
}

// Pass 3 (vector layers): one wave per edge, lane == channel.
// agg[d,:] += (p/ssum[d]) * h[s,:]  -- coalesced 128B gather + 128B f32 atomics.
__global__ void edge_aggregate32(const int* __restrict__ src,
                                 const int* __restrict__ dst,
                                 const float* __restrict__ elog,
                                 const float* __restrict__ ssum,
                                 const float* __restrict__ H,
                                 float* __restrict__ agg) {
  const int lane = threadIdx.x & 31;
  const int e    = (blockIdx.x * blockDim.x + threadIdx.x) >> 5;
  if (e >= ET) return;
  int s, d;
  edge_ids(e, src, dst, s, d);
  const float alpha = elog[e] / ssum[d];
  atomicAdd(&agg[(size_t)d * C + lane], alpha * H[(size_t)s * C + lane]);
}

// Pass 3 (scalar output layer): thread per edge.
__global__ void edge_aggregate1(const int* __restrict__ src,
                                const int* __restrict__ dst,
                                const float* __restrict__ elog,
                                const float* __restrict__ ssum,
                                const float* __restrict__ H1,
                                float* __restrict__ agg1) {
  const int e = blockIdx.x * blockDim.x + threadIdx.x;
  if (e >= ET) return;
  int s, d;
  edge_ids(e, src, dst, s, d);
  atomicAdd(&agg1[d], (elog[e] / ssum[d]) * H1[s]);
}

// Epilogue, input layer: x0 = agg + b; xcur = elu(x0)
__global__ void epi_in(const float* __restrict__ agg, const float* __restrict__ b,
                       float* __restrict__ x0, float* __restrict__ xcur) {
  const int i = blockIdx.x * blockDim.x + threadIdx.x;
  if (i >= NN * C) return;
  const float v = agg[i] + b[i & 31];
  x0[i]   = v;
  xcur[i] = (v > 0.0f) ? v : expm1f(v);
}

// Epilogue, mid layers: xcur = elu(agg + b + x0)
__global__ void epi_mid(const float* __restrict__ agg, const float* __restrict__ b,
                        const float* __restrict__ x0, float* __restrict__ xcur) {
  const int i = blockIdx.x * blockDim.x + threadIdx.x;
  if (i >= NN * C) return;
  const float v = agg[i] + b[i & 31] + x0[i];
  xcur[i] = (v > 0.0f) ? v : expm1f(v);
}

// Output layer: h1[n] = x[n,:].W_out ; asrc/adst = h1 * a_{src,dst}_out (1x1).
__global__ void out_dot(const float* __restrict__ X, const float* __restrict__ Wout,
                        const float* __restrict__ a_s, const float* __restrict__ a_d,
                        float* __restrict__ h1, float* __restrict__ asrc,
                        float* __restrict__ adst) {
  const int lane = threadIdx.x & 31;
  const int n    = (blockIdx.x * blockDim.x + threadIdx.x) >> 5;
  if (n >= NN) return;
  float v = X[(size_t)n * C + lane] * Wout[lane];
#pragma unroll
  for (int off = 16; off > 0; off >>= 1) v += __shfl_xor(v, off, 32);
  if (lane == 0) {
    h1[n]   = v;
    asrc[n] = v * a_s[0];
    adst[n] = v * a_d[0];
  }
}

__global__ void finalize(const float* __restrict__ agg1,
                         const float* __restrict__ b_out,
                         float* __restrict__ out) {
  const int n = blockIdx.x * blockDim.x + threadIdx.x;
  if (n >= NN) return;
  const float v = agg1[n] + b_out[0];
  out[n] = 1.0f / (1.0f + expf(-v));
}

// ---------------------------------------------------------------------------
extern "C" void kernel_launch(void* const* d_in, const int* in_sizes, int n_in,
                              void* d_out, int out_size, void* d_ws, size_t ws_size,
                              hipStream_t stream) {
  (void)in_sizes; (void)n_in; (void)out_size; (void)ws_size;

  const float* X        = (const float*)d_in[0];   // [N,128]
  const int*   eidx     = (const int*)d_in[1];     // [2,E]
  const int*   src      = eidx;
  const int*   dst      = eidx + EE;
  const float* W_in     = (const float*)d_in[2];
  const float* a_src_in = (const float*)d_in[3];
  const float* a_dst_in = (const float*)d_in[4];
  const float* b_in     = (const float*)d_in[5];
  const float* W_mid    = (const float*)d_in[6];
  const float* a_src_md = (const float*)d_in[7];
  const float* a_dst_md = (const float*)d_in[8];
  const float* b_mid    = (const float*)d_in[9];
  const float* W_out    = (const float*)d_in[10];
  const float* a_src_o  = (const float*)d_in[11];
  const float* a_dst_o  = (const float*)d_in[12];
  const float* b_out    = (const float*)d_in[13];

  // Workspace carve-up (~60 MB, fits MI455X 192 MB L2 together with inputs)
  float* ws   = (float*)d_ws;
  float* xcur = ws;                 // [N,32]
  float* x0   = xcur + (size_t)NN * C;
  float* h    = x0   + (size_t)NN * C;   // also h1 for the output layer
  float* agg  = h    + (size_t)NN * C;   // also agg1 for the output layer
  float* asrc = agg  + (size_t)NN * C;
  float* adst = asrc + NN;
  float* mmax = adst + NN;
  float* ssum = mmax + NN;
  float* elog = ssum + NN;          // [E+N]

  const int B = 256;
  const dim3 gGemm((NN / 16 * 32 + B - 1) / B);   // one wave per 16-row tile
  const dim3 gNodeW((NN * 32 + B - 1) / B);       // one wave per node
  const dim3 gNC((NN * C + B - 1) / B);           // elementwise over N*32
  const dim3 gN((NN + B - 1) / B);
  const dim3 gE((ET + B - 1) / B);                // thread per edge
  const dim3 gEW(((size_t)ET * 32 + B - 1) / B);  // wave per edge

  // ---- Input layer (K = 128) ----
  gemm_wmma_f32<128><<<gGemm, B, 0, stream>>>(X, W_in, h);
  node_attn<<<gNodeW, B, 0, stream>>>(h, a_src_in, a_dst_in, asrc, adst);
  init_layer<<<gNC, B, 0, stream>>>(mmax, ssum, agg, NN * C);
  edge_logit_max<<<gE, B, 0, stream>>>(src, dst, asrc, adst, elog, mmax);
  edge_exp_sum<<<gE, B, 0, stream>>>(src, dst, elog, mmax, ssum);
  edge_aggregate32<<<gEW, B, 0, stream>>>(src, dst, elog, ssum, h, agg);
  epi_in<<<gNC, B, 0, stream>>>(agg, b_in, x0, xcur);

  // ---- 6 mid layers, shared weights (K = 32) ----
  for (int l = 0; l < 6; ++l) {
    gemm_wmma_f32<32><<<gGemm, B, 0, stream>>>(xcur, W_mid, h);
    node_attn<<<gNodeW, B, 0, stream>>>(h, a_src_md, a_dst_md, asrc, adst);
    init_layer<<<gNC, B, 0, stream>>>(mmax, ssum, agg, NN * C);
    edge_logit_max<<<gE, B, 0, stream>>>(src, dst, asrc, adst, elog, mmax);
    edge_exp_sum<<<gE, B, 0, stream>>>(src, dst, elog, mmax, ssum);
    edge_aggregate32<<<gEW, B, 0, stream>>>(src, dst, elog, ssum, h, agg);
    epi_mid<<<gNC, B, 0, stream>>>(agg, b_mid, x0, xcur);
  }

  // ---- Output layer (32 -> 1, mean over 1 head == identity, sigmoid) ----
  out_dot<<<gNodeW, B, 0, stream>>>(xcur, W_out, a_src_o, a_dst_o, h, asrc, adst);
  init_layer<<<gN, B, 0, stream>>>(mmax, ssum, agg, NN);
  edge_logit_max<<<gE, B, 0, stream>>>(src, dst, asrc, adst, elog, mmax);
  edge_exp_sum<<<gE, B, 0, stream>>>(src, dst, elog, mmax, ssum);
  edge_aggregate1<<<gE, B, 0, stream>>>(src, dst, elog, ssum, h, agg);
  finalize<<<gN, B, 0, stream>>>(agg, b_out, (float*)d_out);
}